// AttentionManifold_21328807592129
// MI455X (gfx1250) — compile-verified
//
#include <hip/hip_runtime.h>
#include <math.h>

typedef float v2f __attribute__((ext_vector_type(2)));
typedef float v4f __attribute__((ext_vector_type(4)));
typedef float v8f __attribute__((ext_vector_type(8)));

#define LDM 36   // stride for 32x32 LDS matrices: v4f rows 16B-aligned, conflict-free frags
#define LDT 68   // stride for the 32x64 intermediate

// ---------------- WMMA f32 16x16x4 ----------------
__device__ inline v8f wmma4(v2f a, v2f b, v8f c) {
  return __builtin_amdgcn_wmma_f32_16x16x4_f32(
      /*neg_a=*/false, a, /*neg_b=*/false, b,
      /*c_mod=*/(short)0, c, /*reuse_a=*/false, /*reuse_b=*/false);
}

// A-matrix 16x4 fragment (contiguous pair -> b64): lane L -> M=L%16, K=2*(L/16)+reg
__device__ inline v2f load_fragA2(const float* base, int ld, int m0, int k0, int lane) {
  int m = m0 + (lane & 15);
  int k = k0 + ((lane >> 4) << 1);
  return *(const v2f*)&base[m * ld + k];
}
// B-matrix 4x16 fragment, generic (strided, two b32): B[k][n]
__device__ inline v2f load_fragB(const float* base, int ld, int k0, int n0, int lane) {
  int n = n0 + (lane & 15);
  int k = k0 + ((lane >> 4) << 1);
  v2f f; f.x = base[k * ld + n]; f.y = base[(k + 1) * ld + n];
  return f;
}
// B fragment from a SYMMETRIC (or transposed-source) matrix: B[k][n] = src[n][k] -> contiguous b64
__device__ inline v2f load_fragB_sym(const float* base, int ld, int k0, int n0, int lane) {
  int n = n0 + (lane & 15);
  int k = k0 + ((lane >> 4) << 1);
  return *(const v2f*)&base[n * ld + k];
}
// C/D store, generic (column-per-lane, 8x b32)
__device__ inline void store_fragC(float* base, int ld, int m0, int n0, int lane, v8f c) {
  int n = n0 + (lane & 15);
  int mo = m0 + ((lane >> 4) << 3);
#pragma unroll
  for (int r = 0; r < 8; ++r) base[(mo + r) * ld + n] = c[r];
}
// C/D store for SYMMETRIC result: store transposed -> contiguous -> 2x b128
__device__ inline void store_fragC_sym(float* base, int ld, int m0, int n0, int lane, v8f c) {
  int n = n0 + (lane & 15);
  int mo = m0 + ((lane >> 4) << 3);
  v4f lo = {c[0], c[1], c[2], c[3]};
  v4f hi = {c[4], c[5], c[6], c[7]};
  *(v4f*)&base[n * ld + mo]     = lo;
  *(v4f*)&base[n * ld + mo + 4] = hi;
}

// 32x32 = 32x32 * 32x32 for symmetric commuting operands (symmetric result)
__device__ inline void mm32s(float* C, const float* A, const float* B, int lane) {
#pragma unroll
  for (int ti = 0; ti < 2; ++ti)
#pragma unroll
    for (int tj = 0; tj < 2; ++tj) {
      v8f acc = {};
#pragma unroll
      for (int kk = 0; kk < 32; kk += 4)
        acc = wmma4(load_fragA2(A, LDM, ti * 16, kk, lane),
                    load_fragB_sym(B, LDM, kk, tj * 16, lane), acc);
      store_fragC_sym(C, LDM, ti * 16, tj * 16, lane, acc);
    }
}

__device__ inline float wave_sum32(float v) {
#pragma unroll
  for (int off = 16; off; off >>= 1) v += __shfl_xor(v, off, 32);
  return v;
}

// ============ K1: BiMap (W X W^T) + matrix log, one wave per matrix ============
__global__ __launch_bounds__(32) void bimap_log_kernel(
    const float* __restrict__ x, const float* __restrict__ Wq,
    const float* __restrict__ Wk, const float* __restrict__ Wv,
    float* __restrict__ logQ, float* __restrict__ logK, float* __restrict__ logV) {
  const int bm = blockIdx.x;      // 0..511  (b*64+m)
  const int which = blockIdx.y;   // 0:Q 1:K 2:V
  const int lane = threadIdx.x;
  const float* W = (which == 0) ? Wq : (which == 1) ? Wk : Wv;
  float* outp = ((which == 0) ? logQ : (which == 1) ? logK : logV) + bm * 1024;
  const float* X = x + bm * 4096;

  __shared__ float Tb[32 * LDT];       // 32x64 intermediate
  __shared__ float Mb[6][32 * LDM];    // six 32x32 work matrices
  float* A = Mb[0];
  float* Y = Mb[1];
  float* Z = Mb[2];
  float* S1 = Mb[3];
  float* S2 = Mb[4];
  float* S3 = Mb[5];

  // T = W (32x64) * X (64x64); A frags (W rows) contiguous b64 from global
#pragma unroll
  for (int ti = 0; ti < 2; ++ti)
#pragma unroll
    for (int tj = 0; tj < 4; ++tj) {
      v8f acc = {};
#pragma unroll
      for (int kk = 0; kk < 64; kk += 4)
        acc = wmma4(load_fragA2(W, 64, ti * 16, kk, lane),
                    load_fragB(X, 64, kk, tj * 16, lane), acc);
      store_fragC(Tb, LDT, ti * 16, tj * 16, lane, acc);
    }
  // A = T (32x64) * W^T (64x32); B = W^T read row-wise from W (contiguous);
  // result is symmetric -> transposed b128 store
#pragma unroll
  for (int ti = 0; ti < 2; ++ti)
#pragma unroll
    for (int tj = 0; tj < 2; ++tj) {
      v8f acc = {};
#pragma unroll
      for (int kk = 0; kk < 64; kk += 4)
        acc = wmma4(load_fragA2(Tb, LDT, ti * 16, kk, lane),
                    load_fragB_sym(W, 64, kk, tj * 16, lane), acc);
      store_fragC_sym(A, LDM, ti * 16, tj * 16, lane, acc);
    }

  // ---- matrix log: log(A) = log(c) I + 2^3 log(B^{1/8}),  B = A/c ----
  float c = wave_sum32(A[lane * LDM + lane]) * (1.0f / 32.0f);
  float lc = logf(c);
  float ic = 1.0f / c;
#pragma unroll
  for (int cc = 0; cc < 32; cc += 4) {
    v4f v = *(const v4f*)&A[lane * LDM + cc];
    *(v4f*)&A[lane * LDM + cc] = v * ic;
  }

  // three repeated square roots via coupled Newton-Schulz (trace-renormalized)
#pragma unroll 1
  for (int lev = 0; lev < 3; ++lev) {
    float t = wave_sum32(A[lane * LDM + lane]) * (1.0f / 32.0f);
    float it = 1.0f / t;
    float st = sqrtf(t);
    v4f vz = {};
#pragma unroll
    for (int cc = 0; cc < 32; cc += 4) {
      v4f v = *(const v4f*)&A[lane * LDM + cc];
      *(v4f*)&Y[lane * LDM + cc] = v * it;
      *(v4f*)&Z[lane * LDM + cc] = vz;
    }
    Z[lane * LDM + lane] = 1.0f;
#pragma unroll 1
    for (int ns = 0; ns < 8; ++ns) {
      mm32s(S1, Z, Y, lane);                             // S1 = Z*Y
#pragma unroll
      for (int cc = 0; cc < 32; cc += 4) {
        v4f v = *(const v4f*)&S1[lane * LDM + cc];
        *(v4f*)&S1[lane * LDM + cc] = v * -0.5f;         // (3I - S1)/2
      }
      S1[lane * LDM + lane] += 1.5f;
      mm32s(S2, Y, S1, lane);                            // Ynew = Y*T
      mm32s(S3, S1, Z, lane);                            // Znew = T*Z
      float* tmp = Y; Y = S2; S2 = tmp;
      tmp = Z; Z = S3; S3 = tmp;
    }
#pragma unroll
    for (int cc = 0; cc < 32; cc += 4) {
      v4f v = *(const v4f*)&Y[lane * LDM + cc];
      *(v4f*)&A[lane * LDM + cc] = v * st;
    }
  }

  // 4th-order log series: log(I+M) ~ M - M^2/2 + M^3/3 - M^4/4,  M = A - I
#pragma unroll
  for (int cc = 0; cc < 32; cc += 4) {
    v4f v = *(const v4f*)&A[lane * LDM + cc];
    *(v4f*)&S1[lane * LDM + cc] = v;
    *(v4f*)&S2[lane * LDM + cc] = v * -0.25f;
  }
  S1[lane * LDM + lane] += -1.0f;
  S2[lane * LDM + lane] += 0.25f + (1.0f / 3.0f);   // diag of -0.25*(A-I) + (1/3)I
  mm32s(S3, S1, S2, lane);
  S3[lane * LDM + lane] += -0.5f;
  mm32s(S2, S1, S3, lane);
  S2[lane * LDM + lane] += 1.0f;
  mm32s(S3, S1, S2, lane);

  // logP = lc*I + 8 * series ; row-per-lane b128 global stores
#pragma unroll
  for (int cc = 0; cc < 32; cc += 4) {
    v4f v = *(const v4f*)&S3[lane * LDM + cc];
    *(v4f*)&outp[lane * 32 + cc] = v * 8.0f;
  }
  outp[lane * 32 + lane] = 8.0f * S3[lane * LDM + lane] + lc;
}

// ============ K2: energy[b,i,j] = ||logQ[b,j] - logK[b,i]||_F^2 ============
__global__ __launch_bounds__(256) void energy_kernel(
    const float* __restrict__ logQ, const float* __restrict__ logK,
    float* __restrict__ energy) {
  int b = blockIdx.y;
  int pair = blockIdx.x * 256 + threadIdx.x;  // 0..4095
  int i = pair >> 6, j = pair & 63;
  const v4f* q = (const v4f*)(logQ + (b * 64 + j) * 1024);
  const v4f* k = (const v4f*)(logK + (b * 64 + i) * 1024);
  v4f acc = {};
  for (int t = 0; t < 256; ++t) {
    v4f d = q[t] - k[t];
    acc += d * d;
  }
  energy[(b * 64 + i) * 64 + j] = acc[0] + acc[1] + acc[2] + acc[3];
}

// ============ K3: score -> softmax over i -> weight[b,j,i] ============
__global__ __launch_bounds__(64) void softmax_kernel(
    const float* __restrict__ energy, float* __restrict__ weight) {
  int b = blockIdx.y, j = blockIdx.x, i = threadIdx.x;
  float e = energy[(b * 64 + i) * 64 + j];
  float score = 1.0f / (1.0f + log1pf(e));
  __shared__ float red[64];
  red[i] = score;
  __syncthreads();
  for (int off = 32; off; off >>= 1) {
    if (i < off) red[i] = fmaxf(red[i], red[i + off]);
    __syncthreads();
  }
  float mx = red[0];
  __syncthreads();
  float ex = expf(score - mx);
  red[i] = ex;
  __syncthreads();
  for (int off = 32; off; off >>= 1) {
    if (i < off) red[i] += red[i + off];
    __syncthreads();
  }
  weight[(b * 64 + j) * 64 + i] = ex / red[0];
}

// ============ K4: mean_log = sum_i w[j,i] logV[i]; out = expm(mean_log) ============
__global__ __launch_bounds__(32) void mean_exp_kernel(
    const float* __restrict__ weight, const float* __restrict__ logV,
    float* __restrict__ out) {
  int b = blockIdx.y, j = blockIdx.x, lane = threadIdx.x;
  __shared__ float Mb[3][32 * LDM];
  float* S = Mb[0];
  float* G = Mb[1];
  float* P = Mb[2];

  // register accumulation: each lane owns row `lane` of the 32x32 mean (8 x v4f)
  v4f sreg[8];
#pragma unroll
  for (int cc = 0; cc < 8; ++cc) sreg[cc] = (v4f){};
  const float* wrow = weight + (b * 64 + j) * 64;
#pragma unroll 1
  for (int i = 0; i < 64; ++i) {
    float w = wrow[i];
    const v4f* V = (const v4f*)(logV + (b * 64 + i) * 1024 + lane * 32);
#pragma unroll
    for (int cc = 0; cc < 8; ++cc) sreg[cc] += V[cc] * w;
  }
  // S = mean_log / 32 (scaling for exp), spill row to LDS
#pragma unroll
  for (int cc = 0; cc < 8; ++cc)
    *(v4f*)&S[lane * LDM + cc * 4] = sreg[cc] * (1.0f / 32.0f);

  // Horner Taylor-6: G = I + S/6
#pragma unroll
  for (int cc = 0; cc < 32; cc += 4) {
    v4f v = *(const v4f*)&S[lane * LDM + cc];
    *(v4f*)&G[lane * LDM + cc] = v * (1.0f / 6.0f);
  }
  G[lane * LDM + lane] += 1.0f;
#pragma unroll 1
  for (int k = 5; k >= 1; --k) {
    mm32s(P, S, G, lane);
    float ik = 1.0f / (float)k;
#pragma unroll
    for (int cc = 0; cc < 32; cc += 4) {
      v4f v = *(const v4f*)&P[lane * LDM + cc];
      *(v4f*)&G[lane * LDM + cc] = v * ik;
    }
    G[lane * LDM + lane] += 1.0f;
  }
  // square 5 times: exp(S*32) = G^(2^5)
#pragma unroll 1
  for (int sq = 0; sq < 5; ++sq) {
    mm32s(P, G, G, lane);
#pragma unroll
    for (int cc = 0; cc < 32; cc += 4)
      *(v4f*)&G[lane * LDM + cc] = *(const v4f*)&P[lane * LDM + cc];
  }

  float* o = out + (b * 64 + j) * 1024;
#pragma unroll
  for (int cc = 0; cc < 32; cc += 4)
    *(v4f*)&o[lane * 32 + cc] = *(const v4f*)&G[lane * LDM + cc];
}

// ============ launch ============
extern "C" void kernel_launch(void* const* d_in, const int* in_sizes, int n_in,
                              void* d_out, int out_size, void* d_ws, size_t ws_size,
                              hipStream_t stream) {
  (void)in_sizes; (void)n_in; (void)out_size; (void)ws_size;
  const float* x  = (const float*)d_in[0];   // [8,64,64,64]
  const float* Wq = (const float*)d_in[1];   // [32,64]
  const float* Wk = (const float*)d_in[2];
  const float* Wv = (const float*)d_in[3];
  float* ws = (float*)d_ws;

  const int NMAT = 8 * 64;                  // 512
  float* logQ = ws;                         // 512*1024
  float* logK = logQ + NMAT * 1024;
  float* logV = logK + NMAT * 1024;
  float* energy = logV + NMAT * 1024;       // 8*64*64
  float* weight = energy + 8 * 64 * 64;     // 8*64*64

  bimap_log_kernel<<<dim3(NMAT, 3), 32, 0, stream>>>(x, Wq, Wk, Wv, logQ, logK, logV);
  energy_kernel<<<dim3(16, 8), 256, 0, stream>>>(logQ, logK, energy);
  softmax_kernel<<<dim3(64, 8), 64, 0, stream>>>(energy, weight);
  mean_exp_kernel<<<dim3(64, 8), 32, 0, stream>>>(weight, logV, (float*)d_out);
}